// ASymmqGPS_42365557407995
// MI455X (gfx1250) — compile-verified
//
#include <hip/hip_runtime.h>

// ---------------------------------------------------------------------------
// ASymmqGPS log-psi: 4096 independent 64x64 complex slogdets + tiny combine.
// One 256-thread (8xwave32) workgroup per matrix; split-complex FP32 LU with
// partial pivoting in LDS; blocked (NB=16) right-looking LU whose trailing
// Schur updates run on v_wmma_f32_16x16x4_f32 (4 real WMMAs per complex tile).
// Log-magnitude / phase of the diagonal accumulated in FP64.
// ---------------------------------------------------------------------------

#define NMAT   64
#define LSTR   65            // LDS row stride (floats), padded vs 64 banks
#define NB     16
#define DNUM   8
#define BNUM   256
#define LSITES 128

typedef float v2f __attribute__((ext_vector_type(2)));
typedef float v8f __attribute__((ext_vector_type(8)));

// ---------------------------------------------------------------------------
// Kernel A: occupancy bits -> electron site indices. 512 tasks (b, spin).
// ---------------------------------------------------------------------------
__global__ void occ_kernel(const int* __restrict__ x, int* __restrict__ yIdx) {
  int t = blockIdx.x * blockDim.x + threadIdx.x;
  if (t >= 2 * BNUM) return;
  int b = t >> 1;
  int spin = t & 1;
  int mask = spin ? 2 : 1;
  int* out = yIdx + (spin * BNUM + b) * NMAT;
  int cnt = 0;
  for (int l = 0; l < LSITES; ++l) {
    if (x[b * LSITES + l] & mask) {
      if (cnt < NMAT) out[cnt] = l;
      ++cnt;
    }
  }
}

// ---------------------------------------------------------------------------
// Kernel B: one workgroup per (spin, d, b) matrix. Gather -> LU -> logdet.
// ---------------------------------------------------------------------------
__global__ __launch_bounds__(256) void slogdet_kernel(
    const double* __restrict__ orb_up,   // (D, L, 64) complex128 interleaved
    const double* __restrict__ orb_dn,
    const int* __restrict__ yIdx,        // [2][B][64]
    double* __restrict__ logdets)        // [2][D][B][2] = (logmag, phase)
{
  __shared__ float Ar[NMAT * LSTR];
  __shared__ float Ai[NMAT * LSTR];
  __shared__ float pvVal[NMAT];
  __shared__ int   pvIdx[NMAT];
  __shared__ double s_logmag, s_phase;

  const int tid  = threadIdx.x;
  const int lane = tid & 31;
  const int wave = tid >> 5;

  const int gid  = blockIdx.x;             // [0, 2*D*B)
  const int spin = gid >> 11;              // D*B = 2048
  const int rest = gid & 2047;
  const int d    = rest >> 8;
  const int b    = rest & 255;

  const double* orb = spin ? orb_dn : orb_up;
  const int* y = yIdx + (spin * BNUM + b) * NMAT;

  // Gather electron-indexed rows, complex128 -> split fp32 planes in LDS.
  for (int e = tid; e < NMAT * NMAT; e += 256) {
    int r = e >> 6, n = e & 63;
    int site = y[r];
    const double* src = orb + 2 * (size_t)((d * LSITES + site) * NMAT + n);
    Ar[r * LSTR + n] = (float)src[0];
    Ai[r * LSTR + n] = (float)src[1];
  }
  if (tid == 0) { s_logmag = 0.0; s_phase = 0.0; }
  __syncthreads();

  for (int kb = 0; kb < NMAT; kb += NB) {
    // ----- panel factorization: columns kb..kb+15, pivot over rows k..63 ---
    for (int j = 0; j < NB; ++j) {
      const int k = kb + j;
      if (tid < NMAT) {
        float v = -1.0f;
        if (tid >= k) {
          float re = Ar[tid * LSTR + k];
          float im = Ai[tid * LSTR + k];
          v = re * re + im * im;
        }
        pvVal[tid] = v;
        pvIdx[tid] = tid;
      }
      __syncthreads();
      for (int s = 32; s > 0; s >>= 1) {      // argmax tree-reduce
        if (tid < s) {
          if (pvVal[tid + s] > pvVal[tid]) {
            pvVal[tid] = pvVal[tid + s];
            pvIdx[tid] = pvIdx[tid + s];
          }
        }
        __syncthreads();
      }
      const int p = pvIdx[0];
      if (p != k && tid < NMAT) {             // full-row swap
        float tr = Ar[p * LSTR + tid];
        Ar[p * LSTR + tid] = Ar[k * LSTR + tid];
        Ar[k * LSTR + tid] = tr;
        float ti = Ai[p * LSTR + tid];
        Ai[p * LSTR + tid] = Ai[k * LSTR + tid];
        Ai[k * LSTR + tid] = ti;
      }
      __syncthreads();
      const float pr = Ar[k * LSTR + k];
      const float pi = Ai[k * LSTR + k];
      if (tid == 0) {                         // fp64 logdet accumulation
        double m2 = (double)pr * (double)pr + (double)pi * (double)pi;
        s_logmag += 0.5 * log(m2);
        double ph = atan2((double)pi, (double)pr);
        if (p != k) ph += 3.14159265358979323846;  // row swap: det *= -1
        s_phase += ph;
      }
      const float inv2 = 1.0f / (pr * pr + pi * pi);
      const float ivr  = pr * inv2;
      const float ivi  = -pi * inv2;
      if (tid < NMAT - (k + 1)) {             // scale sub-column -> L
        int r = k + 1 + tid;
        float xr = Ar[r * LSTR + k], xi = Ai[r * LSTR + k];
        Ar[r * LSTR + k] = xr * ivr - xi * ivi;
        Ai[r * LSTR + k] = xr * ivi + xi * ivr;
      }
      __syncthreads();
      const int nc = kb + NB - (k + 1);       // rank-1 update inside panel
      const int nr = NMAT - (k + 1);
      for (int idx = tid; idx < nr * nc; idx += 256) {
        int r = k + 1 + idx / nc;
        int c = k + 1 + idx % nc;
        float lr = Ar[r * LSTR + k], li = Ai[r * LSTR + k];
        float ur = Ar[k * LSTR + c], ui = Ai[k * LSTR + c];
        Ar[r * LSTR + c] -= lr * ur - li * ui;
        Ai[r * LSTR + c] -= lr * ui + li * ur;
      }
      __syncthreads();
    }

    const int rem = NMAT - (kb + NB);
    if (rem <= 0) break;

    // ----- U12 = L11^{-1} * A12: forward substitution, one thread/column ---
    if (tid < rem) {
      int c = kb + NB + tid;
      for (int i = 1; i < NB; ++i) {
        float sr = Ar[(kb + i) * LSTR + c];
        float si = Ai[(kb + i) * LSTR + c];
        for (int jj = 0; jj < i; ++jj) {
          float lr = Ar[(kb + i) * LSTR + kb + jj];
          float li = Ai[(kb + i) * LSTR + kb + jj];
          float ur = Ar[(kb + jj) * LSTR + c];
          float ui = Ai[(kb + jj) * LSTR + c];
          sr -= lr * ur - li * ui;
          si -= lr * ui + li * ur;
        }
        Ar[(kb + i) * LSTR + c] = sr;
        Ai[(kb + i) * LSTR + c] = si;
      }
    }
    __syncthreads();

    // ----- A22 -= L21 * U12 via v_wmma_f32_16x16x4_f32 (complex = 4 WMMA) --
    const int nt = rem >> 4;                  // 16x16 tiles per dim: 3,2,1
    for (int t = wave; t < nt * nt; t += 8) {
      int tm = t / nt, tn = t % nt;
      int mBase = kb + NB + tm * 16;
      int nBase = kb + NB + tn * 16;

      // C/D layout: c[r] -> (M = r or r+8, N = lane&15)
      int cN  = nBase + (lane & 15);
      int cM0 = mBase + ((lane >> 4) << 3);
      v8f cr, ci;
#pragma unroll
      for (int r = 0; r < 8; ++r) {
        cr[r] = Ar[(cM0 + r) * LSTR + cN];
        ci[r] = Ai[(cM0 + r) * LSTR + cN];
      }

#pragma unroll
      for (int kk = 0; kk < NB; kk += 4) {
        // A 16x4: lanes 0-15 -> K={0,1}, lanes 16-31 -> K={2,3}
        int aM = mBase + (lane & 15);
        int aK = kb + kk + ((lane >> 4) << 1);
        float a0r = Ar[aM * LSTR + aK], a1r = Ar[aM * LSTR + aK + 1];
        float a0i = Ai[aM * LSTR + aK], a1i = Ai[aM * LSTR + aK + 1];
        v2f arN; arN[0] = -a0r; arN[1] = -a1r;   // -Re(L21)
        v2f aiP; aiP[0] =  a0i; aiP[1] =  a1i;   // +Im(L21)
        v2f aiN; aiN[0] = -a0i; aiN[1] = -a1i;   // -Im(L21)
        // B 4x16: lanes 0-15 -> K={0,1}, lanes 16-31 -> K={2,3}, N = lane&15
        int bN = nBase + (lane & 15);
        int bK = kb + kk + ((lane >> 4) << 1);
        v2f br, bi;
        br[0] = Ar[bK * LSTR + bN]; br[1] = Ar[(bK + 1) * LSTR + bN];
        bi[0] = Ai[bK * LSTR + bN]; bi[1] = Ai[(bK + 1) * LSTR + bN];

        // Cr += (-Ar)Br + (Ai)Bi ; Ci += (-Ar)Bi + (-Ai)Br
        cr = __builtin_amdgcn_wmma_f32_16x16x4_f32(false, arN, false, br,
                                                   (short)0, cr, false, false);
        cr = __builtin_amdgcn_wmma_f32_16x16x4_f32(false, aiP, false, bi,
                                                   (short)0, cr, false, false);
        ci = __builtin_amdgcn_wmma_f32_16x16x4_f32(false, arN, false, bi,
                                                   (short)0, ci, false, false);
        ci = __builtin_amdgcn_wmma_f32_16x16x4_f32(false, aiN, false, br,
                                                   (short)0, ci, false, false);
      }

#pragma unroll
      for (int r = 0; r < 8; ++r) {
        Ar[(cM0 + r) * LSTR + cN] = cr[r];
        Ai[(cM0 + r) * LSTR + cN] = ci[r];
      }
    }
    __syncthreads();
  }

  if (tid == 0) {
    double* o = logdets + 2 * (size_t)((spin * DNUM + d) * BNUM + b);
    o[0] = s_logmag;
    o[1] = s_phase;
  }
}

// ---------------------------------------------------------------------------
// Kernel C: log(sinh(sum_d exp(logdet_up + logdet_dn))) per batch, complex128.
// ---------------------------------------------------------------------------
__global__ void combine_kernel(const double* __restrict__ logdets,
                               double* __restrict__ out) {
  int b = blockIdx.x * blockDim.x + threadIdx.x;
  if (b >= BNUM) return;
  double sr = 0.0, si = 0.0;
  for (int d = 0; d < DNUM; ++d) {
    const double* u = logdets + 2 * (size_t)((0 * DNUM + d) * BNUM + b);
    const double* v = logdets + 2 * (size_t)((1 * DNUM + d) * BNUM + b);
    double re = u[0] + v[0];
    double ph = u[1] + v[1];
    double m = exp(re);
    sr += m * cos(ph);
    si += m * sin(ph);
  }
  // w = sinh(sr + i*si); out = log(w)
  double wr = sinh(sr) * cos(si);
  double wi = cosh(sr) * sin(si);
  out[2 * b + 0] = 0.5 * log(wr * wr + wi * wi);
  out[2 * b + 1] = atan2(wi, wr);
}

// ---------------------------------------------------------------------------
extern "C" void kernel_launch(void* const* d_in, const int* in_sizes, int n_in,
                              void* d_out, int out_size, void* d_ws,
                              size_t ws_size, hipStream_t stream) {
  (void)in_sizes; (void)n_in; (void)out_size; (void)ws_size;
  const int*    x      = (const int*)d_in[0];
  const double* orb_up = (const double*)d_in[1];
  const double* orb_dn = (const double*)d_in[2];

  int*    yIdx    = (int*)d_ws;                                  // 128 KB
  double* logdets = (double*)((char*)d_ws + 2 * BNUM * NMAT * sizeof(int));

  occ_kernel<<<2, 256, 0, stream>>>(x, yIdx);
  slogdet_kernel<<<2 * DNUM * BNUM, 256, 0, stream>>>(orb_up, orb_dn, yIdx,
                                                      logdets);
  combine_kernel<<<1, 256, 0, stream>>>(logdets, (double*)d_out);
}